// BatchAllTtripletLoss_54331336294803
// MI455X (gfx1250) — compile-verified
//
#include <hip/hip_runtime.h>
#include <hip/hip_bf16.h>

typedef __attribute__((ext_vector_type(2))) float v2f;
typedef __attribute__((ext_vector_type(8))) float v8f;

constexpr int BN  = 640;   // batch
constexpr int DIM = 512;   // embedding dim
constexpr float kEPS    = 1e-8f;
constexpr float kMARGIN = 0.3f;
constexpr int NT = BN / 16;  // 40 tiles per side

// ---------------------------------------------------------------------------
// Kernel 1: Gram matrix G = X * X^T via V_WMMA_F32_16X16X4_F32.
// One wave32 per 16x16 output tile; K-loop of 128 WMMA issues per tile.
// ---------------------------------------------------------------------------
__global__ __launch_bounds__(32)
void gram_wmma_kernel(const float* __restrict__ X, float* __restrict__ G) {
  const int tile = blockIdx.x;          // 0 .. 40*40-1
  const int ti = tile / NT;
  const int tj = tile % NT;
  const int lane = threadIdx.x;         // 0..31 (wave32)

  // A fragment (16x4, f32): lanes 0-15 -> M=lane, K = k0+{0,1}
  //                         lanes16-31 -> M=lane-16, K = k0+{2,3}
  const int m      = ti * 16 + (lane & 15);
  const int n      = tj * 16 + (lane & 15);
  const int khalf  = (lane >> 4) << 1;  // 0 or 2
  const float* arow = X + (size_t)m * DIM + khalf;
  // B = X^T : B[k][n] = X[n][k]  -> same striding pattern on the column row.
  const float* brow = X + (size_t)n * DIM + khalf;

  v8f c = {0.f, 0.f, 0.f, 0.f, 0.f, 0.f, 0.f, 0.f};

#pragma unroll 8
  for (int k0 = 0; k0 < DIM; k0 += 4) {
    v2f a = *(const v2f*)(arow + k0);   // 8-byte aligned (khalf even)
    v2f b = *(const v2f*)(brow + k0);
    // (neg_a, A, neg_b, B, c_mod, C, reuse_a, reuse_b)
    c = __builtin_amdgcn_wmma_f32_16x16x4_f32(
        false, a, false, b, (short)0, c, false, false);
  }

  // D layout: VGPR v -> M = ti*16 + v + (lane>=16 ? 8 : 0), N = tj*16 + (lane&15)
  const int mbase = ti * 16 + ((lane >> 4) << 3);
#pragma unroll
  for (int v = 0; v < 8; ++v) {
    G[(size_t)(mbase + v) * BN + n] = c[v];
  }
}

// ---------------------------------------------------------------------------
// Kernel 2: Euclidean distance matrix from Gram (exact reference semantics).
// ---------------------------------------------------------------------------
__global__ __launch_bounds__(256)
void dmat_kernel(const float* __restrict__ G, float* __restrict__ Dm) {
  int idx = blockIdx.x * blockDim.x + threadIdx.x;
  if (idx >= BN * BN) return;
  int i = idx / BN;
  int j = idx - i * BN;
  float v = G[(size_t)i * BN + i] - 2.0f * G[idx] + G[(size_t)j * BN + j];
  v = fmaxf(v, 0.0f);
  float zm = (v == 0.0f) ? 1.0f : 0.0f;
  Dm[idx] = sqrtf(v + zm * kEPS) * (1.0f - zm);
}

// ---------------------------------------------------------------------------
// Kernel 3: batch-all triplet reduction. One block per anchor i.
// Row i of dmat + labels staged in LDS; inner k-loop is an LDS broadcast.
// valid(i,j,k): i!=j, lab[j]==lab[i], lab[k]!=lab[i] (implies k!=i, k!=j).
// ---------------------------------------------------------------------------
__global__ __launch_bounds__(256)
void triplet_kernel(const float* __restrict__ Dm, const int* __restrict__ labels,
                    float* __restrict__ partials) {
  __shared__ float srow[BN];
  __shared__ int   slab[BN];
  __shared__ float rsum[256];
  __shared__ float rcnt[256];

  const int i   = blockIdx.x;
  const int tid = threadIdx.x;

  for (int t = tid; t < BN; t += 256) {
    srow[t] = Dm[(size_t)i * BN + t];
    slab[t] = labels[t];
  }
  __syncthreads();

  const int li = slab[i];
  float sum = 0.0f;
  float cnt = 0.0f;

  for (int j = tid; j < BN; j += 256) {
    if (j == i || slab[j] != li) continue;
    const float dij = srow[j] + kMARGIN;
    for (int k = 0; k < BN; ++k) {       // all lanes read same srow[k]: broadcast
      if (slab[k] == li) continue;
      float t = dij - srow[k];
      if (t > 0.0f)  sum += t;
      if (t > kEPS)  cnt += 1.0f;
    }
  }

  rsum[tid] = sum;
  rcnt[tid] = cnt;
  __syncthreads();
  for (int s = 128; s > 0; s >>= 1) {
    if (tid < s) { rsum[tid] += rsum[tid + s]; rcnt[tid] += rcnt[tid + s]; }
    __syncthreads();
  }
  if (tid == 0) {
    partials[i]      = rsum[0];
    partials[BN + i] = rcnt[0];
  }
}

// ---------------------------------------------------------------------------
// Kernel 4: final deterministic reduction -> scalar loss.
// ---------------------------------------------------------------------------
__global__ __launch_bounds__(256)
void final_kernel(const float* __restrict__ partials, float* __restrict__ out) {
  __shared__ float rsum[256];
  __shared__ float rcnt[256];
  const int tid = threadIdx.x;
  float s = 0.0f, c = 0.0f;
  for (int t = tid; t < BN; t += 256) {
    s += partials[t];
    c += partials[BN + t];
  }
  rsum[tid] = s;
  rcnt[tid] = c;
  __syncthreads();
  for (int st = 128; st > 0; st >>= 1) {
    if (tid < st) { rsum[tid] += rsum[tid + st]; rcnt[tid] += rcnt[tid + st]; }
    __syncthreads();
  }
  if (tid == 0) out[0] = rsum[0] / (rcnt[0] + kEPS);
}

// ---------------------------------------------------------------------------
extern "C" void kernel_launch(void* const* d_in, const int* in_sizes, int n_in,
                              void* d_out, int out_size, void* d_ws, size_t ws_size,
                              hipStream_t stream) {
  const float* X      = (const float*)d_in[0];   // (640, 512) fp32
  const int*   labels = (const int*)d_in[1];     // (640,) int
  float*       out    = (float*)d_out;           // scalar fp32

  float* G        = (float*)d_ws;                   // 640*640 f32
  float* Dm       = G + (size_t)BN * BN;            // 640*640 f32
  float* partials = Dm + (size_t)BN * BN;           // 2*640 f32

  gram_wmma_kernel<<<NT * NT, 32, 0, stream>>>(X, G);
  dmat_kernel<<<(BN * BN + 255) / 256, 256, 0, stream>>>(G, Dm);
  triplet_kernel<<<BN, 256, 0, stream>>>(Dm, labels, partials);
  final_kernel<<<1, 256, 0, stream>>>(partials, out);
}